// SGC_74483322847409
// MI455X (gfx1250) — compile-verified
//
#include <hip/hip_runtime.h>

#define D_FEAT 128

typedef __attribute__((ext_vector_type(2))) float v2f;
typedef __attribute__((ext_vector_type(8))) float v8f;

// --- Stage 1: degree / normalization --------------------------------------
// Self-loops add 1 to every node's in-degree, so start norm[] at 1.0.
__global__ void SGC_init_norm(float* __restrict__ norm, int n) {
    int i = blockIdx.x * blockDim.x + threadIdx.x;
    if (i < n) norm[i] = 1.0f;
}

__global__ void SGC_count_deg(const int* __restrict__ ei, float* __restrict__ norm, int e) {
    int i = blockIdx.x * blockDim.x + threadIdx.x;
    if (i < e) atomicAdd(&norm[ei[(size_t)e + i]], 1.0f);  // col = ei[1][i]
}

// deg >= 1 always (self loop), so no zero guard needed. ALPHA = 0.5 makes
// both left/right factors deg^-1/2.
__global__ void SGC_finalize_norm(float* __restrict__ norm, int n) {
    int i = blockIdx.x * blockDim.x + threadIdx.x;
    if (i < n) norm[i] = rsqrtf(norm[i]);
}

// --- Stage 2: SpMM hop -----------------------------------------------------
// Initialize xout with the self-loop contribution: xout[i] = norm[i]^2 * xin[i].
// (Avoids a separate zeroing pass.)
__global__ void SGC_self_loop(const float* __restrict__ xin,
                              const float* __restrict__ norm,
                              float* __restrict__ xout, int n) {
    int i = blockIdx.x * blockDim.x + threadIdx.x;   // over n * 32 float4s
    int total = n * (D_FEAT / 4);
    if (i >= total) return;
    int node = i / (D_FEAT / 4);
    float w = norm[node];
    w = w * w;
    float4 v = ((const float4*)xin)[i];
    ((float4*)xout)[i] = make_float4(v.x * w, v.y * w, v.z * w, v.w * w);
}

// One wave32 per edge; lane L owns features [4L, 4L+4). 4 f32 atomics/lane.
__global__ void SGC_spmm_edges(const float* __restrict__ xin,
                               const int* __restrict__ ei,
                               const float* __restrict__ norm,
                               float* __restrict__ xout, int e_count) {
    int lane = threadIdx.x & 31;
    int edge = (int)((blockIdx.x * (size_t)blockDim.x + threadIdx.x) >> 5);
    if (edge >= e_count) return;
    int row = ei[edge];
    int col = ei[(size_t)e_count + edge];
    float w = norm[row] * norm[col];
    float4 v = ((const float4*)(xin + (size_t)col * D_FEAT))[lane];
    float* dst = xout + (size_t)row * D_FEAT + lane * 4;
    atomicAdd(dst + 0, w * v.x);
    atomicAdd(dst + 1, w * v.y);
    atomicAdd(dst + 2, w * v.z);
    atomicAdd(dst + 3, w * v.w);
}

// --- Stage 3: out = h @ W^T + b via V_WMMA_F32_16X16X4_F32 -----------------
// Block = 256 threads = 8 waves; wave w owns output column tile n0 = 16*w,
// block owns row tile m0 = 16*blockIdx.x. K loop: 128 in steps of 4 -> 32 WMMA.
//
// A (16x4, MxK) layout (ISA 7.12.2): lane L holds M = L&15; VGPR0 holds
//   K = k0 + (L>>4)*2, VGPR1 the next K. -> float2 load from h.
// B (4x16, KxN): lane L holds N = L&15; VGPR g holds K = k0 + g + (L>>4)*2.
//   B[k][n] = W[n][k] (W is row-major D_OUT x D_IN). -> float2 load from W row.
// C/D (16x16): VGPR g, lanes 0-15 -> M = g, lanes 16-31 -> M = g+8, N = L&15.
__global__ void __launch_bounds__(256)
SGC_gemm_wmma(const float* __restrict__ h, const float* __restrict__ W,
              const float* __restrict__ bias, float* __restrict__ out, int n_nodes) {
    const int wave  = threadIdx.x >> 5;   // 0..7 : column tile
    const int lane  = threadIdx.x & 31;
    const int lhalf = lane >> 4;          // 0 | 1
    const int l15   = lane & 15;
    const int m0 = blockIdx.x * 16;
    const int n0 = wave * 16;
    if (m0 >= n_nodes) return;            // uniform per block -> EXEC stays full

    float bv = bias[n0 + l15];            // bias depends only on N = l15
    v8f c = {bv, bv, bv, bv, bv, bv, bv, bv};

    int arow_idx = m0 + l15;
    if (arow_idx >= n_nodes) arow_idx = n_nodes - 1;   // tail clamp (reads only)
    const float* arow = h + (size_t)arow_idx * D_FEAT + lhalf * 2;
    const float* brow = W + (size_t)(n0 + l15) * D_FEAT + lhalf * 2;

#pragma unroll
    for (int k0 = 0; k0 < D_FEAT; k0 += 4) {
        v2f a = *(const v2f*)(arow + k0);
        v2f bb = *(const v2f*)(brow + k0);
        // 8 args: (neg_a, A, neg_b, B, c_mod, C, reuse_a, reuse_b)
        c = __builtin_amdgcn_wmma_f32_16x16x4_f32(false, a, false, bb,
                                                  (short)0, c, false, false);
    }

    const int mbase = m0 + lhalf * 8;
    float* obase = out + (size_t)mbase * D_FEAT + n0 + l15;
#pragma unroll
    for (int g = 0; g < 8; ++g) {
        if (mbase + g < n_nodes) obase[(size_t)g * D_FEAT] = c[g];
    }
}

extern "C" void kernel_launch(void* const* d_in, const int* in_sizes, int n_in,
                              void* d_out, int out_size, void* d_ws, size_t ws_size,
                              hipStream_t stream) {
    const float* x    = (const float*)d_in[0];
    const int*   ei   = (const int*)d_in[1];   // edge_index [2, E] row-major, int32
    const float* W    = (const float*)d_in[2];
    const float* bias = (const float*)d_in[3];
    float*       out  = (float*)d_out;

    const int n = in_sizes[0] / D_FEAT;
    const int e = in_sizes[1] / 2;

    char* ws = (char*)d_ws;
    size_t norm_bytes = (((size_t)n * sizeof(float)) + 255) & ~(size_t)255;
    size_t buf_bytes  = (size_t)n * D_FEAT * sizeof(float);
    float* norm = (float*)ws;
    float* bufA = (float*)(ws + norm_bytes);
    float* bufB = (float*)(ws + norm_bytes + buf_bytes);

    const int T = 256;
    // degree + normalization
    SGC_init_norm<<<(n + T - 1) / T, T, 0, stream>>>(norm, n);
    SGC_count_deg<<<(e + T - 1) / T, T, 0, stream>>>(ei, norm, e);
    SGC_finalize_norm<<<(n + T - 1) / T, T, 0, stream>>>(norm, n);

    const int tot4 = n * (D_FEAT / 4);
    const size_t edge_threads = (size_t)e * 32;
    // hop 1: x -> bufA
    SGC_self_loop<<<(tot4 + T - 1) / T, T, 0, stream>>>(x, norm, bufA, n);
    SGC_spmm_edges<<<(int)((edge_threads + T - 1) / T), T, 0, stream>>>(x, ei, norm, bufA, e);
    // hop 2: bufA -> bufB
    SGC_self_loop<<<(tot4 + T - 1) / T, T, 0, stream>>>(bufA, norm, bufB, n);
    SGC_spmm_edges<<<(int)((edge_threads + T - 1) / T), T, 0, stream>>>(bufA, ei, norm, bufB, e);
    // dense tail on the WMMA pipe (fp32 precision preserved)
    SGC_gemm_wmma<<<(n + 15) / 16, T, 0, stream>>>(bufB, W, bias, out, n);
}